// GLIF3_47579647705289
// MI455X (gfx1250) — compile-verified
//
#include <hip/hip_runtime.h>
#include <hip/hip_bf16.h>

// ---------------------------------------------------------------------------
// GLIF3 neuron scan for MI455X (gfx1250).
// Memory-bound sequential scan: state lives in VGPRs for all 1000 steps,
// I_ext streamed via gfx1250 async global->LDS loads (32-deep ring buffer,
// tracked with ASYNCcnt), outputs streamed with non-temporal stores.
// ---------------------------------------------------------------------------

namespace {

constexpr int   kB      = 8;
constexpr int   kT      = 1000;
constexpr int   kN      = 4096;
constexpr float kDT     = 0.002f;
constexpr int   kBlock  = 256;
constexpr int   kStages = 32;          // async pipeline depth (power of 2, < 63)

static_assert((kStages & (kStages - 1)) == 0, "kStages must be power of two");
static_assert(kT > kStages, "pipeline deeper than sequence");

typedef __attribute__((address_space(1))) int global_i32;
typedef __attribute__((address_space(3))) int lds_i32;

// ---- CDNA5 async global->LDS copy (ASYNCcnt-tracked) ----------------------
__device__ __forceinline__ void async_load_f32(const float* gsrc, float* lds_dst) {
#if __has_builtin(__builtin_amdgcn_global_load_async_to_lds_b32)
  __builtin_amdgcn_global_load_async_to_lds_b32(
      (global_i32*)gsrc, (lds_i32*)lds_dst,
      /*imm offset=*/0, /*cpol=*/0);
#else
  unsigned lds_off = (unsigned)(unsigned long long)(lds_i32*)lds_dst;
  asm volatile("global_load_async_to_lds_b32 %0, %1, off"
               :: "v"(lds_off), "v"(gsrc)
               : "memory");
#endif
}

// ---- split dependency-counter waits ---------------------------------------
#if __has_builtin(__builtin_amdgcn_s_wait_asynccnt)
#define WAIT_ASYNCCNT(n)                                   \
  do {                                                     \
    __builtin_amdgcn_s_wait_asynccnt((short)(n));          \
    asm volatile("" ::: "memory");                         \
  } while (0)
#else
#define WAIT_ASYNCCNT(n) asm volatile("s_wait_asynccnt %0" :: "n"(n) : "memory")
#endif

#define WAIT_DSCNT0() asm volatile("s_wait_dscnt 0" ::: "memory")

// ---- one GLIF3 timestep (matches reference op order exactly) --------------
__device__ __forceinline__ void glif3_step(
    float cur,
    float& V, float& a0, float& a1, float& frozen,
    float kv, float rvkv, float vrest, float vreset, float vth, float slen,
    float k0, float k1, float r0, float r1, float A0, float A1,
    float& S_ret, float& V_ret)
{
  float sfd = fmaxf(kDT + frozen, 0.0f);          // spike-free fraction of dt
  a0 = a0 * (1.0f - sfd * k0);                    // ASC decay (used below)
  a1 = a1 * (1.0f - sfd * k1);
  float Vp = V - sfd * kv * (V - vrest) + sfd * (cur + a0 + a1) * rvkv;
  float S  = ((Vp > vth) && (frozen == 0.0f)) ? 1.0f : 0.0f;
  a0 = a0 + S * (a0 * r0 + A0);                   // after-spike kick
  a1 = a1 + S * (a1 * r1 + A1);
  frozen = fminf(frozen - S * slen + kDT, 0.0f);  // refractory clock
  V = (S == 1.0f) ? vreset : Vp;
  S_ret = S;
  V_ret = V;
}

__global__ __launch_bounds__(kBlock, 2)
void glif3_scan_kernel(const float* __restrict__ I_ext,
                       const float* __restrict__ K_V,
                       const float* __restrict__ R_V_K_V,
                       const float* __restrict__ V_rest,
                       const float* __restrict__ V_reset,
                       const float* __restrict__ V_th,
                       const float* __restrict__ spike_len,
                       const float* __restrict__ K_Ij,
                       const float* __restrict__ R_Ij,
                       const float* __restrict__ A_Ij,
                       float* __restrict__ S_out,
                       float* __restrict__ V_out)
{
  __shared__ float stage[kStages * kBlock];       // 32 KB ring buffer

  const int tid = threadIdx.x;
  const int g   = blockIdx.x * kBlock + tid;      // flattened (b, n)
  const int b   = g >> 12;                        // g / kN  (kN == 4096)
  const int n   = g & (kN - 1);                   // g % kN

  // per-neuron parameters: registers for the whole scan
  const float kv    = K_V[n];
  const float rvkv  = R_V_K_V[n];
  const float vrest = V_rest[n];
  const float vrst  = V_reset[n];
  const float vth   = V_th[n];
  const float slen  = spike_len[n];
  const float k0 = K_Ij[n],      k1 = K_Ij[kN + n];
  const float r0 = R_Ij[n],      r1 = R_Ij[kN + n];
  const float A0 = A_Ij[n],      A1 = A_Ij[kN + n];

  // recurrent state
  float V = vrest, a0 = 0.0f, a1 = 0.0f, frozen = 0.0f;

  const size_t base  = (size_t)b * kT * kN + n;   // element (b, t=0, n)
  const float* gbase = I_ext + base;
  float*       sbase = S_out + base;
  float*       vbase = V_out + base;

  // ---- prime the async pipeline: stages 0..31 in flight -------------------
#pragma unroll
  for (int s = 0; s < kStages; ++s)
    async_load_f32(gbase + (size_t)s * kN, &stage[s * kBlock + tid]);

  // ---- steady state: wait oldest, consume, refill slot, store -------------
  for (int t = 0; t < kT - kStages; ++t) {
    const int slot = (t & (kStages - 1)) * kBlock + tid;

    WAIT_ASYNCCNT(kStages - 1);                   // oldest async load landed
    const float cur = stage[slot];
    WAIT_DSCNT0();                                // slot read before refill
    async_load_f32(gbase + (size_t)(t + kStages) * kN, &stage[slot]);

    float S, Vn;
    glif3_step(cur, V, a0, a1, frozen,
               kv, rvkv, vrest, vrst, vth, slen, k0, k1, r0, r1, A0, A1,
               S, Vn);

    const size_t off = (size_t)t * kN;            // streaming, write-once:
    __builtin_nontemporal_store(S,  sbase + off); // don't rinse through cache
    __builtin_nontemporal_store(Vn, vbase + off);
  }

  // ---- drain: everything outstanding is for the last kStages steps --------
  WAIT_ASYNCCNT(0);
  for (int t = kT - kStages; t < kT; ++t) {
    const float cur = stage[(t & (kStages - 1)) * kBlock + tid];

    float S, Vn;
    glif3_step(cur, V, a0, a1, frozen,
               kv, rvkv, vrest, vrst, vth, slen, k0, k1, r0, r1, A0, A1,
               S, Vn);

    const size_t off = (size_t)t * kN;
    __builtin_nontemporal_store(S,  sbase + off);
    __builtin_nontemporal_store(Vn, vbase + off);
  }
}

} // anonymous namespace

extern "C" void kernel_launch(void* const* d_in, const int* in_sizes, int n_in,
                              void* d_out, int out_size, void* d_ws, size_t ws_size,
                              hipStream_t stream) {
  (void)in_sizes; (void)n_in; (void)out_size; (void)d_ws; (void)ws_size;

  const float* I_ext     = (const float*)d_in[0];
  const float* K_V       = (const float*)d_in[1];
  const float* R_V_K_V   = (const float*)d_in[2];
  const float* V_rest    = (const float*)d_in[3];
  const float* V_reset   = (const float*)d_in[4];
  const float* V_th      = (const float*)d_in[5];
  const float* spike_len = (const float*)d_in[6];
  const float* K_Ij      = (const float*)d_in[7];
  const float* R_Ij      = (const float*)d_in[8];
  const float* A_Ij      = (const float*)d_in[9];

  float* S_out = (float*)d_out;                               // [B,T,N]
  float* V_out = (float*)d_out + (size_t)kB * kT * kN;        // [B,T,N]

  const dim3 grid((kB * kN) / kBlock);   // 32768 threads = one per (b, n)
  const dim3 block(kBlock);
  glif3_scan_kernel<<<grid, block, 0, stream>>>(
      I_ext, K_V, R_V_K_V, V_rest, V_reset, V_th, spike_len,
      K_Ij, R_Ij, A_Ij, S_out, V_out);
}